// GAT_24137716203798
// MI455X (gfx1250) — compile-verified
//
#include <hip/hip_runtime.h>
#include <hip/hip_bf16.h>

typedef __attribute__((ext_vector_type(2))) float v2f;
typedef __attribute__((ext_vector_type(8))) float v8f;

#define NEG_BIG (-9.0e15f)
#define POS_CAP (3.0e38f)

__device__ __forceinline__ float lrelu(float x) { return x > 0.0f ? x : 0.2f * x; }
__device__ __forceinline__ float elu(float x)   { return x > 0.0f ? x : __expf(x) - 1.0f; }

__device__ __forceinline__ float wredmax(float v) {
  #pragma unroll
  for (int o = 16; o > 0; o >>= 1) v = fmaxf(v, __shfl_xor(v, o, 32));
  return v;
}
__device__ __forceinline__ float wredsum(float v) {
  #pragma unroll
  for (int o = 16; o > 0; o >>= 1) v += __shfl_xor(v, o, 32);
  return v;
}

// Branch-free masked score: scores are O(10), so clamping against a selected
// constant reproduces (adj>0 ? score : NEG_BIG) while forcing the score math
// to be unconditional (select is between two immediates -> single v_cndmask,
// nothing for the compiler to sink into an EXEC-masked branch).
__device__ __forceinline__ float masked_score(int a, float score) {
  const float cap = (a > 0) ? POS_CAP : NEG_BIG;
  return fminf(score, cap);
}

// ---------------------------------------------------------------------------
// Generic f32 WMMA GEMM: C[M,N] = act(A[M,K] @ op(B) + bias)
// A row-major (lda). BT=false: B row-major [K,Nfull] (ldb). BT=true: B is
// [Nfull,K] row-major (ldb) and we use B^T. 128 threads = 4 waves; each wave
// computes a 16x16 tile of a 32x32 block. K multiple of 64; M,N multiples of 32.
// ---------------------------------------------------------------------------
template <bool BT, bool ELUACT, bool HASBIAS>
__global__ void gemm_f32_wmma(const float* __restrict__ A, int lda,
                              const float* __restrict__ B, int ldb,
                              const float* __restrict__ bias,
                              float* __restrict__ C, int ldc, int K) {
  __shared__ __align__(16) float As[32 * 68];  // stride 68: float4-aligned, conflict-free frag reads
  __shared__ float Bs[64 * 33];                // stride 33: conflict-free row reads
  const int tid  = threadIdx.x;   // 128
  const int wid  = tid >> 5;
  const int lane = tid & 31;
  const int m16  = lane & 15;
  const int hi   = lane >> 4;
  const int tm   = (wid >> 1) * 16;
  const int tn   = (wid & 1) * 16;
  const int rowbase = blockIdx.x * 32;
  const int nb      = blockIdx.y * 32;

  v8f acc = {};

  for (int kb = 0; kb < K; kb += 64) {
    // stage A tile: 32 rows x 64 k (coalesced float4 loads)
    #pragma unroll
    for (int i = 0; i < 4; ++i) {
      int idx = tid + 128 * i;
      int r = idx >> 4, c4 = (idx & 15) << 2;
      float4 v = *(const float4*)(A + (size_t)(rowbase + r) * lda + kb + c4);
      *(float4*)(&As[r * 68 + c4]) = v;
    }
    // stage B tile: Bs[k][n], 64 x 32
    if (!BT) {
      #pragma unroll
      for (int i = 0; i < 4; ++i) {
        int idx = tid + 128 * i;
        int kr = idx >> 3, c4 = (idx & 7) << 2;
        float4 v = *(const float4*)(B + (size_t)(kb + kr) * ldb + nb + c4);
        Bs[kr * 33 + c4 + 0] = v.x;
        Bs[kr * 33 + c4 + 1] = v.y;
        Bs[kr * 33 + c4 + 2] = v.z;
        Bs[kr * 33 + c4 + 3] = v.w;
      }
    } else {
      #pragma unroll
      for (int i = 0; i < 4; ++i) {
        int idx = tid + 128 * i;
        int n = idx >> 4, k4 = (idx & 15) << 2;
        float4 v = *(const float4*)(B + (size_t)(nb + n) * ldb + kb + k4);
        Bs[(k4 + 0) * 33 + n] = v.x;
        Bs[(k4 + 1) * 33 + n] = v.y;
        Bs[(k4 + 2) * 33 + n] = v.z;
        Bs[(k4 + 3) * 33 + n] = v.w;
      }
    }
    __syncthreads();

    #pragma unroll
    for (int kk = 0; kk < 64; kk += 4) {
      // A frag 16x4: elem e, half hi -> K = kk + 2*hi + e (consecutive: ds_load_b64)
      float2 av = *(const float2*)&As[(tm + m16) * 68 + kk + 2 * hi];
      v2f a; a.x = av.x; a.y = av.y;
      // B frag 4x16: same K mapping, N = tn + m16
      v2f b;
      b.x = Bs[(kk + 2 * hi + 0) * 33 + tn + m16];
      b.y = Bs[(kk + 2 * hi + 1) * 33 + tn + m16];
      acc = __builtin_amdgcn_wmma_f32_16x16x4_f32(false, a, false, b, (short)0, acc,
                                                  false, false);
    }
    __syncthreads();
  }

  const int col = nb + tn + m16;
  const float bv = HASBIAS ? bias[col] : 0.0f;
  #pragma unroll
  for (int v = 0; v < 8; ++v) {
    float x = acc[v] + bv;  // D layout: VGPR v -> row v (lanes 0-15) / v+8 (lanes 16-31)
    if (ELUACT) x = elu(x);
    C[(size_t)(rowbase + tm + v + 8 * hi) * ldc + col] = x;
  }
}

// ---------------------------------------------------------------------------
// s1[n] = Wh[n,:64] . a[0:64],  s2[n] = Wh[n,:64] . a[64:128]
// 256 threads = 8 waves, one node per wave.
// ---------------------------------------------------------------------------
__global__ void score_kernel(const float* __restrict__ Wh, int ld,
                             const float* __restrict__ a,
                             float* __restrict__ s1, float* __restrict__ s2) {
  const int wid = threadIdx.x >> 5, lane = threadIdx.x & 31;
  const int n = blockIdx.x * 8 + wid;
  const float* row = Wh + (size_t)n * ld;
  const float w1 = row[lane], w2 = row[lane + 32];
  float v1 = w1 * a[lane] + w2 * a[lane + 32];
  float v2 = w1 * a[64 + lane] + w2 * a[96 + lane];
  v1 = wredsum(v1);
  v2 = wredsum(v2);
  if (lane == 0) { s1[n] = v1; s2[n] = v2; }
}

// ---------------------------------------------------------------------------
// Fused masked-softmax attention + PV matmul + ELU for a 16-row tile.
// 512 threads = 16 waves. Scores are rank-1 (s1[i]+s2[j]) through lrelu, so
// row max = lrelu(s1[i] + max_{adj} s2[j]) by monotonicity. Two wide (b128)
// adjacency scans (max, Z), then a streamed PV pass with f32 WMMA on a
// normalized probability tile staged in LDS. All masking is branch-free.
// ---------------------------------------------------------------------------
__global__ void attn_kernel(const float* __restrict__ V, int ldv,
                            const float* __restrict__ s1, const float* __restrict__ s2,
                            const int* __restrict__ adj, int N,
                            float* __restrict__ out, int ldo) {
  __shared__ __align__(16) float Pl[16 * 66];  // P tile, stride 66 -> conflict-free b64 frag reads
  __shared__ float RED[16 * 256];              // cross-strip reduction
  __shared__ float maxs[16], invZ[16];

  const int tid  = threadIdx.x;  // 512
  const int wid  = tid >> 5;
  const int lane = tid & 31;
  const int i0   = blockIdx.x * 16;
  const int r    = wid;  // scan phases: one wave per row
  const int m16  = lane & 15;
  const int hi   = lane >> 4;

  const int* arow = adj + (size_t)(i0 + r) * N;
  const float s1v = s1[i0 + r];

  // ---- pass 1: masked row max of s2 (b128 stream, branch-free) ----
  float mx = -3.0e38f;
  for (int j4 = lane * 4; j4 < N; j4 += 128) {
    const int4   a4 = *(const int4*)(arow + j4);
    const float4 s4 = *(const float4*)(s2 + j4);
    mx = fmaxf(mx, fminf(s4.x, a4.x > 0 ? POS_CAP : -3.0e38f));
    mx = fmaxf(mx, fminf(s4.y, a4.y > 0 ? POS_CAP : -3.0e38f));
    mx = fmaxf(mx, fminf(s4.z, a4.z > 0 ? POS_CAP : -3.0e38f));
    mx = fmaxf(mx, fminf(s4.w, a4.w > 0 ? POS_CAP : -3.0e38f));
  }
  mx = wredmax(mx);
  const float mr = (mx > -1.0e38f) ? lrelu(s1v + mx) : NEG_BIG;
  if (lane == 0) maxs[r] = mr;

  // ---- pass 2: Z = sum exp(score - max) (branch-free) ----
  float z = 0.0f;
  for (int j4 = lane * 4; j4 < N; j4 += 128) {
    const int4   a4 = *(const int4*)(arow + j4);
    const float4 s4 = *(const float4*)(s2 + j4);
    const float e0 = masked_score(a4.x, lrelu(s1v + s4.x));
    const float e1 = masked_score(a4.y, lrelu(s1v + s4.y));
    const float e2 = masked_score(a4.z, lrelu(s1v + s4.z));
    const float e3 = masked_score(a4.w, lrelu(s1v + s4.w));
    z += __expf(e0 - mr) + __expf(e1 - mr) + __expf(e2 - mr) + __expf(e3 - mr);
  }
  z = wredsum(z);
  if (lane == 0) invZ[r] = 1.0f / z;
  __syncthreads();

  // per-row constants held in registers across the whole chunk loop
  // (avoids per-iteration ds_loads the compiler cannot hoist past barriers)
  const float iz = invZ[r];

  // ---- pass 3: streamed P@V with WMMA ----
  const int c  = wid & 3;   // column tile (16 cols each)
  const int q  = wid >> 2;  // j-strip within chunk
  const int col0 = c * 16;
  v8f acc = {};

  for (int jb = 0; jb < N; jb += 64) {
    // build normalized probability tile P[16 x 64]: lane handles 2 consecutive j
    {
      const int jl = lane * 2;
      const int2   a2 = *(const int2*)(arow + jb + jl);
      const float2 s4 = *(const float2*)(s2 + jb + jl);
      const float e0 = masked_score(a2.x, lrelu(s1v + s4.x));
      const float e1 = masked_score(a2.y, lrelu(s1v + s4.y));
      float2 p;
      p.x = __expf(e0 - mr) * iz;
      p.y = __expf(e1 - mr) * iz;
      *(float2*)&Pl[r * 66 + jl] = p;  // (r*66 + 2*lane) even -> 8B aligned
      if (jb + 64 < N) __builtin_prefetch(arow + jb + 64 + jl, 0, 1);
    }
    __syncthreads();

    // each wave: 16 j's (strip q), 16 output cols (tile c), 4x K=4 WMMA
    {
      const int jj = q * 16;
      #pragma unroll
      for (int k0 = 0; k0 < 16; k0 += 4) {
        float2 av = *(const float2*)&Pl[m16 * 66 + jj + k0 + 2 * hi];
        v2f a; a.x = av.x; a.y = av.y;
        const int j0 = jb + jj + k0 + 2 * hi;
        v2f b;
        b.x = V[(size_t)(j0 + 0) * ldv + col0 + m16];
        b.y = V[(size_t)(j0 + 1) * ldv + col0 + m16];
        acc = __builtin_amdgcn_wmma_f32_16x16x4_f32(false, a, false, b, (short)0, acc,
                                                    false, false);
      }
    }
    __syncthreads();
  }

  // reduce the 4 j-strips per column tile, ELU, store
  #pragma unroll
  for (int v = 0; v < 8; ++v) RED[wid * 256 + lane * 8 + v] = acc[v];
  __syncthreads();
  if (wid < 4) {
    #pragma unroll
    for (int v = 0; v < 8; ++v) {
      float x = RED[(0 * 4 + wid) * 256 + lane * 8 + v] +
                RED[(1 * 4 + wid) * 256 + lane * 8 + v] +
                RED[(2 * 4 + wid) * 256 + lane * 8 + v] +
                RED[(3 * 4 + wid) * 256 + lane * 8 + v];
      x = elu(x);
      out[(size_t)(i0 + v + 8 * hi) * ldo + wid * 16 + m16] = x;
    }
  }
}

__global__ void add_kernel(const float* __restrict__ a, const float* __restrict__ b,
                           float* __restrict__ o, int n) {
  const int i = blockIdx.x * 256 + threadIdx.x;
  if (i < n) o[i] = a[i] + b[i];
}

// ---------------------------------------------------------------------------

extern "C" void kernel_launch(void* const* d_in, const int* in_sizes, int n_in,
                              void* d_out, int out_size, void* d_ws, size_t ws_size,
                              hipStream_t stream) {
  (void)in_sizes; (void)n_in; (void)out_size; (void)ws_size;

  const float* x        = (const float*)d_in[0];
  const int*   adj      = (const int*)d_in[1];
  const float* w_heads  = (const float*)d_in[2];  // [4,128,64]
  const float* a_heads  = (const float*)d_in[3];  // [4,128,1]
  const float* w_out    = (const float*)d_in[4];  // [256,64]
  const float* a_out    = (const float*)d_in[5];  // [128,1]
  const float* lin_w    = (const float*)d_in[6];  // [64,128]
  const float* lin_b    = (const float*)d_in[7];  // [64]
  const float* outlin_w = (const float*)d_in[8];  // [64,64]
  const float* outlin_b = (const float*)d_in[9];  // [64]
  float* out = (float*)d_out;                     // [4096,64]

  constexpr int N = 4096, FIN = 128, H = 64, NH = 4;

  // workspace layout (floats)
  float* ws     = (float*)d_ws;
  float* Wh_all = ws;                             // [N,256]
  float* x_res  = Wh_all + (size_t)N * 256;       // [N,64]
  float* Wh_out = x_res  + (size_t)N * H;         // [N,64]
  float* hcat   = Wh_out + (size_t)N * H;         // [N,256]
  float* att_o  = hcat   + (size_t)N * 256;       // [N,64]
  float* tmp    = att_o  + (size_t)N * H;         // [N,64]
  float* s1h    = tmp    + (size_t)N * H;         // [4,N]
  float* s2h    = s1h    + (size_t)NH * N;        // [4,N]
  float* s1o    = s2h    + (size_t)NH * N;        // [N]
  float* s2o    = s1o    + N;                     // [N]

  const dim3 gblk(128);

  // 1) Wh per head: Wh_all[:, h*64:(h+1)*64] = x @ w_heads[h]
  for (int h = 0; h < NH; ++h)
    gemm_f32_wmma<false, false, false><<<dim3(N / 32, H / 32), gblk, 0, stream>>>(
        x, FIN, w_heads + (size_t)h * FIN * H, H, nullptr, Wh_all + h * H, 256, FIN);

  // 2) x_residual = x @ lin_w^T + lin_b
  gemm_f32_wmma<true, false, true><<<dim3(N / 32, H / 32), gblk, 0, stream>>>(
      x, FIN, lin_w, FIN, lin_b, x_res, H, FIN);

  // 3) attention score projections per head
  for (int h = 0; h < NH; ++h)
    score_kernel<<<N / 8, 256, 0, stream>>>(Wh_all + h * H, 256,
                                            a_heads + (size_t)h * 2 * H,
                                            s1h + (size_t)h * N, s2h + (size_t)h * N);

  // 4) per-head fused attention -> hcat[:, h*64:(h+1)*64]
  for (int h = 0; h < NH; ++h)
    attn_kernel<<<N / 16, 512, 0, stream>>>(Wh_all + h * H, 256,
                                            s1h + (size_t)h * N, s2h + (size_t)h * N,
                                            adj, N, hcat + h * H, 256);

  // 5) output GAT layer: Wh_out = hcat @ w_out
  gemm_f32_wmma<false, false, false><<<dim3(N / 32, H / 32), gblk, 0, stream>>>(
      hcat, 256, w_out, H, nullptr, Wh_out, H, 256);

  // 6) output layer scores
  score_kernel<<<N / 8, 256, 0, stream>>>(Wh_out, H, a_out, s1o, s2o);

  // 7) output attention
  attn_kernel<<<N / 16, 512, 0, stream>>>(Wh_out, H, s1o, s2o, adj, N, att_o, H);

  // 8) residual add
  add_kernel<<<(N * H) / 256, 256, 0, stream>>>(x_res, att_o, tmp, N * H);

  // 9) final: elu((x_res + att) @ outlin_w^T + outlin_b)
  gemm_f32_wmma<true, true, true><<<dim3(N / 32, H / 32), gblk, 0, stream>>>(
      tmp, H, outlin_w, H, outlin_b, out, H, H);
}